// GIN_45028437131840
// MI455X (gfx1250) — compile-verified
//
#include <hip/hip_runtime.h>
#include <math.h>

// ---------------------------------------------------------------------------
// GIN forward for MI455X (gfx1250, wave32).
// Memory-bound on edge scatter (~1.3 GB @ 23.3 TB/s); GEMMs (1.4 GFLOP f32)
// use V_WMMA_F32_16X16X4_F32, one wave per 16-row block, 8x A-fragment reuse,
// branch-free inner loop with immediate-offset loads.
// ---------------------------------------------------------------------------

#define N_NODES 10000
#define N_EDGES 640000
#define DIM 128
#define N_CLASSES 40
#define NCP 48                      // N_CLASSES padded to 3 tiles of 16
#define HN (N_NODES * DIM)          // 1,280,000 floats
#define BN_BLOCKS 100               // 100 rows per block
#define BN_EPS 1e-5f
#define MTILES (N_NODES / 16)       // 625

typedef __attribute__((ext_vector_type(2))) float v2f;
typedef __attribute__((ext_vector_type(8))) float v8f;

// ---------------------------------------------------------------- copy h->agg
__global__ void copy_f4(float* __restrict__ dst, const float* __restrict__ src) {
    int i = blockIdx.x * blockDim.x + threadIdx.x;   // 320,000 float4
    ((float4*)dst)[i] = ((const float4*)src)[i];
}

// ------------------------------------------------------- edge scatter-add
// one wave32 per edge; lane moves float4 (coalesced b128 load), 4 f32 atomics
__global__ void scatter_add(float* __restrict__ agg, const float* __restrict__ h,
                            const int* __restrict__ ei) {
    int lane = threadIdx.x & 31;
    int e = (blockIdx.x * blockDim.x + threadIdx.x) >> 5;
    if (e >= N_EDGES) return;
    int s = ei[e];
    int d = ei[N_EDGES + e];
    float4 v = ((const float4*)(h + (size_t)s * DIM))[lane];
    float* ap = agg + (size_t)d * DIM + lane * 4;
    unsafeAtomicAdd(ap + 0, v.x);
    unsafeAtomicAdd(ap + 1, v.y);
    unsafeAtomicAdd(ap + 2, v.z);
    unsafeAtomicAdd(ap + 3, v.w);
}

// ------------------------------------------------------- WMMA f32 GEMM
// C[16*Mtiles, NT*16] = A[., 128] @ B[128, NT*16] + bias, optional ReLU.
// One wave per 16-row M block; NT accumulators reuse each A fragment NT times.
// LDB compile-time => all loads are base-ptr + 24-bit immediate offset.
template <int NT>
__global__ void wmma_gemm_f32(const float* __restrict__ A, const float* __restrict__ B,
                              const float* __restrict__ bias, float* __restrict__ C,
                              int relu) {
    constexpr int LDB = NT * 16;
    int lane = threadIdx.x & 31;
    int wave = threadIdx.x >> 5;
    int tm = blockIdx.x * (blockDim.x >> 5) + wave;
    if (tm >= MTILES) return;                  // wave-uniform exit
    int g  = lane >> 4;                        // lane half: 0 or 1
    int ml = lane & 15;
    int m  = tm * 16 + ml;                     // A-fragment row
    const float* aptr = A + (size_t)m * DIM + 2 * g;  // A[m][k0 + 2g (+1)]
    const float* bptr = B + 2 * g * LDB + ml;         // B[k0+2g (+1)][t*16+ml]

    v8f acc[NT];
#pragma unroll
    for (int t = 0; t < NT; ++t) acc[t] = (v8f){};

#pragma unroll 4
    for (int k0 = 0; k0 < DIM; k0 += 4) {
        v2f a;
        a.x = aptr[k0];
        a.y = aptr[k0 + 1];
#pragma unroll
        for (int t = 0; t < NT; ++t) {
            v2f b;
            b.x = bptr[k0 * LDB + t * 16];
            b.y = bptr[(k0 + 1) * LDB + t * 16];
            acc[t] = __builtin_amdgcn_wmma_f32_16x16x4_f32(
                /*neg_a=*/false, a, /*neg_b=*/false, b,
                /*c_mod=*/(short)0, acc[t], /*reuse_a=*/false, /*reuse_b=*/false);
        }
    }

#pragma unroll
    for (int t = 0; t < NT; ++t) {
        float bv = bias[t * 16 + ml];
#pragma unroll
        for (int r = 0; r < 8; ++r) {
            int mr = tm * 16 + r + 8 * g;      // C/D: VGPR r -> M = r + 8*laneHalf
            float v = acc[t][r] + bv;
            if (relu) v = fmaxf(v, 0.0f);
            C[(size_t)mr * LDB + t * 16 + ml] = v;
        }
    }
}

// ------------------------------------------------------- pad Wfc/bfc to 48 cols
__global__ void pad_fc(const float* __restrict__ Wfc, const float* __restrict__ bfc,
                       float* __restrict__ wpad, float* __restrict__ bpad) {
    int i = blockIdx.x * blockDim.x + threadIdx.x;
    if (i < DIM * NCP) {
        int k = i / NCP, n = i % NCP;
        wpad[i] = (n < N_CLASSES) ? Wfc[k * N_CLASSES + n] : 0.0f;
    } else if (i < DIM * NCP + NCP) {
        int n = i - DIM * NCP;
        bpad[n] = (n < N_CLASSES) ? bfc[n] : 0.0f;
    }
}

// ------------------------------------------------------- BN: partial sums
__global__ void bn_partial(const float* __restrict__ h, float* __restrict__ part) {
    int col = threadIdx.x;                     // 128 threads
    int r0 = blockIdx.x * (N_NODES / BN_BLOCKS);
    float s = 0.f, s2 = 0.f;
    for (int r = 0; r < N_NODES / BN_BLOCKS; ++r) {
        float v = h[(size_t)(r0 + r) * DIM + col];
        s += v; s2 += v * v;
    }
    part[blockIdx.x * 256 + col]       = s;
    part[blockIdx.x * 256 + 128 + col] = s2;
}

// ------------------------------------------------------- BN: finalize mu/rstd
__global__ void bn_finalize(const float* __restrict__ part,
                            float* __restrict__ mu, float* __restrict__ rstd) {
    int col = threadIdx.x;                     // 128 threads, 1 block
    float s = 0.f, s2 = 0.f;
    for (int p = 0; p < BN_BLOCKS; ++p) {
        s  += part[p * 256 + col];
        s2 += part[p * 256 + 128 + col];
    }
    float m = s / (float)N_NODES;
    float var = s2 / (float)N_NODES - m * m;
    mu[col]   = m;
    rstd[col] = rsqrtf(var + BN_EPS);
}

// ------------------------------------------------------- BN apply + ReLU
__global__ void bn_apply_relu(float* __restrict__ h, const float* __restrict__ mu,
                              const float* __restrict__ rstd,
                              const float* __restrict__ gamma,
                              const float* __restrict__ beta) {
    int i = blockIdx.x * blockDim.x + threadIdx.x;   // 1,280,000 elems
    int col = i & (DIM - 1);
    float v = (h[i] - mu[col]) * rstd[col] * gamma[col] + beta[col];
    h[i] = fmaxf(v, 0.0f);
}

// ------------------------------------------------------- log_softmax (40 of 48)
__global__ void log_softmax40(const float* __restrict__ logits, float* __restrict__ out) {
    int lane = threadIdx.x & 31;
    int row = (blockIdx.x * blockDim.x + threadIdx.x) >> 5;
    if (row >= N_NODES) return;
    const float* lp = logits + (size_t)row * NCP;
    float v0 = lp[lane];                                   // lane 0..31 < 40
    float v1 = (lane + 32 < N_CLASSES) ? lp[lane + 32] : -3.0e38f;
    float mx = fmaxf(v0, v1);
#pragma unroll
    for (int o = 16; o > 0; o >>= 1) mx = fmaxf(mx, __shfl_xor(mx, o, 32));
    float e = expf(v0 - mx) + ((lane + 32 < N_CLASSES) ? expf(v1 - mx) : 0.0f);
#pragma unroll
    for (int o = 16; o > 0; o >>= 1) e += __shfl_xor(e, o, 32);
    float lse = logf(e) + mx;
    out[(size_t)row * N_CLASSES + lane] = v0 - lse;
    if (lane + 32 < N_CLASSES)
        out[(size_t)row * N_CLASSES + lane + 32] = v1 - lse;
}

// ---------------------------------------------------------------------------
extern "C" void kernel_launch(void* const* d_in, const int* in_sizes, int n_in,
                              void* d_out, int out_size, void* d_ws, size_t ws_size,
                              hipStream_t stream) {
    const float* x   = (const float*)d_in[0];
    const int*   ei  = (const int*)d_in[1];
    const float* W1a = (const float*)d_in[2];  const float* b1a = (const float*)d_in[3];
    const float* g1  = (const float*)d_in[4];  const float* be1 = (const float*)d_in[5];
    const float* W1b = (const float*)d_in[6];  const float* b1b = (const float*)d_in[7];
    const float* W2a = (const float*)d_in[8];  const float* b2a = (const float*)d_in[9];
    const float* g2  = (const float*)d_in[10]; const float* be2 = (const float*)d_in[11];
    const float* W2b = (const float*)d_in[12]; const float* b2b = (const float*)d_in[13];
    const float* Wfc = (const float*)d_in[14]; const float* bfc = (const float*)d_in[15];

    float* ws = (float*)d_ws;                  // offsets in floats
    float* agg    = ws;                        // 1,280,000
    float* hbuf   = ws + 1310720;              // 1,280,000
    float* tbuf   = ws + 2621440;              // 1,280,000
    float* part   = ws + 3932160;              // 25,600
    float* mu     = ws + 3964928;              // 128
    float* rstd   = ws + 3965056;              // 128
    float* wpad   = ws + 3965184;              // 6,144  (128 x 48)
    float* bpad   = ws + 3971328;              // 48
    float* logits = ws + 3971392;              // 480,000 (10000 x 48)

    const int scat_blocks = (N_EDGES * 32) / 256;          // 80,000
    const int gemm_blocks = (MTILES + 3) / 4;              // 157 (4 waves/blk)

    // ---- layer 1 : agg = x + scatter_add(x)
    copy_f4<<<HN / 4 / 256, 256, 0, stream>>>(agg, x);
    scatter_add<<<scat_blocks, 256, 0, stream>>>(agg, x, ei);
    wmma_gemm_f32<8><<<gemm_blocks, 128, 0, stream>>>(agg, W1a, b1a, hbuf, 0);
    bn_partial<<<BN_BLOCKS, 128, 0, stream>>>(hbuf, part);
    bn_finalize<<<1, 128, 0, stream>>>(part, mu, rstd);
    bn_apply_relu<<<HN / 256, 256, 0, stream>>>(hbuf, mu, rstd, g1, be1);
    wmma_gemm_f32<8><<<gemm_blocks, 128, 0, stream>>>(hbuf, W1b, b1b, tbuf, 1);

    // ---- layer 2 : agg = h + scatter_add(h)
    copy_f4<<<HN / 4 / 256, 256, 0, stream>>>(agg, tbuf);
    scatter_add<<<scat_blocks, 256, 0, stream>>>(agg, tbuf, ei);
    wmma_gemm_f32<8><<<gemm_blocks, 128, 0, stream>>>(agg, W2a, b2a, hbuf, 0);
    bn_partial<<<BN_BLOCKS, 128, 0, stream>>>(hbuf, part);
    bn_finalize<<<1, 128, 0, stream>>>(part, mu, rstd);
    bn_apply_relu<<<HN / 256, 256, 0, stream>>>(hbuf, mu, rstd, g2, be2);
    wmma_gemm_f32<8><<<gemm_blocks, 128, 0, stream>>>(hbuf, W2b, b2b, agg, 1);

    // ---- classifier (padded to 48 cols) + log_softmax
    pad_fc<<<(DIM * NCP + NCP + 255) / 256, 256, 0, stream>>>(Wfc, bfc, wpad, bpad);
    wmma_gemm_f32<3><<<gemm_blocks, 128, 0, stream>>>(agg, wpad, bpad, logits, 0);
    log_softmax40<<<(N_NODES * 32) / 256, 256, 0, stream>>>(logits, (float*)d_out);
}